// DarkChannelPrior_62036507623976
// MI455X (gfx1250) — compile-verified
//
#include <hip/hip_runtime.h>

// Dark Channel Prior: dark = min(B,G,R); out = 15x15 box-average (pad 7), f32.
// Fused single kernel: channel-min + separable box filter in LDS.
// Vertical 15-tap reduction done as banded matmul on the CDNA5 matrix pipe
// using V_WMMA_F32_16X16X4_F32 (exact f32, matches reference precision).
//
// Round 2: tile height 32 -> 64 (halo redundancy 1.6x -> 1.35x, ~45 MB less
// read traffic). LDS grows to ~90 KB/block; gfx1250 WGP has 320 KB, so
// 3 blocks/WGP (24 waves) still gives good latency hiding.

typedef float v2f __attribute__((ext_vector_type(2)));
typedef float v8f __attribute__((ext_vector_type(8)));

#define TX    128            // tile width  (output cols per block)
#define TY    64             // tile height (output rows per block)
#define HALO  14             // 2*7 halo for 15-tap filter
#define DW    (TX + HALO)    // 142 dark cols in LDS
#define DH    (TY + HALO)    // 78  dark/hsum rows in LDS
#define DP    144            // dark LDS pitch (floats)
#define HP    144            // hsum LDS pitch: 2*HP mod 64 = 32 -> lane halves hit disjoint banks
#define IMG_W 1024
#define IMG_H 1024

__global__ __launch_bounds__(256)
void dark_channel_box15(const float* __restrict__ img, float* __restrict__ out)
{
    __shared__ float sDark[DH * DP];   // min over channels, halo-extended  (~44.9 KB)
    __shared__ float sH[DH * HP];      // horizontal 15-sums (TX cols valid) (~44.9 KB)

    const int tid   = threadIdx.x;
    const int tileX = blockIdx.x;      // 0..(1024/TX - 1)
    const int tileY = blockIdx.y;      // 0..(1024/TY - 1)
    const int b     = blockIdx.z;      // batch 0..15
    const int R0 = tileY * TY;
    const int C0 = tileX * TX;

    const size_t plane = (size_t)IMG_W * IMG_H;
    const float* p0 = img + (size_t)b * 3 * plane;
    const float* p1 = p0 + plane;
    const float* p2 = p0 + 2 * plane;

    // ---- Phase 1: dark = min3(channels), zero outside image (= conv zero pad)
    for (int idx = tid; idx < DH * DW; idx += 256) {
        int r  = idx / DW;
        int c  = idx - r * DW;
        int gy = R0 + r - 7;
        int gx = C0 + c - 7;
        float v = 0.0f;
        if ((unsigned)gy < (unsigned)IMG_H && (unsigned)gx < (unsigned)IMG_W) {
            size_t o = (size_t)gy * IMG_W + gx;
            v = fminf(p0[o], fminf(p1[o], p2[o]));
        }
        sDark[r * DP + c] = v;
    }
    __syncthreads();

    // ---- Phase 2: horizontal 15-sum, sH[r][c] = sum_{j=0..14} sDark[r][c+j]
    // lanes sweep consecutive cols of one row -> conflict-free LDS
    for (int idx = tid; idx < DH * TX; idx += 256) {   // 78*128/256 = 39 iters
        int r = idx >> 7;                               // TX == 128
        int c = idx & (TX - 1);
        const float* row = &sDark[r * DP + c];
        float s = 0.0f;
        #pragma unroll
        for (int j = 0; j < 15; ++j) s += row[j];
        sH[r * HP + c] = s;
    }
    __syncthreads();

    // ---- Phase 3: vertical 15-sum as banded matmul on the matrix pipe.
    // O(16x16) = A(16x32 band of ones) x H(32x16), via 8 x wmma_f32_16x16x4_f32.
    // A layout (ISA 16x4 f32 A): lanes 0-15 M=lane, lanes 16-31 M=lane-16;
    //   VGPR0 holds K = base+{0 | 2}, VGPR1 holds K = base+{1 | 3} per lane half.
    // B layout mirrored: lane holds N = lane%16, same K-per-half pattern.
    const int wave = tid >> 5;
    const int lane = tid & 31;
    const int half = lane >> 4;        // 0: lanes 0-15, 1: lanes 16-31
    const int n    = lane & 15;        // M for A, N for B/D
    const float inv225 = 1.0f / 225.0f;

    for (int st = wave; st < (TY / 16) * (TX / 16); st += 8) {  // 32 subtiles, 4/wave
        const int sy = st >> 3;          // subtile row 0..3
        const int sx = st & 7;           // subtile col 0..7
        const int rbase = sy * 16;       // first hsum row of the 32-row K window
        const int colw  = sx * 16 + n;

        v8f acc = {0.f, 0.f, 0.f, 0.f, 0.f, 0.f, 0.f, 0.f};
        #pragma unroll
        for (int t = 0; t < 8; ++t) {
            const int k0 = t * 4 + half * 2;
            const int k1 = k0 + 1;
            // Band: output row m (= n) sums hsum rows m..m+14 of the window.
            v2f a;
            a.x = (k0 >= n && k0 <= n + 14) ? 1.0f : 0.0f;
            a.y = (k1 >= n && k1 <= n + 14) ? 1.0f : 0.0f;
            // Clamp rows where A==0 anyway (avoid NaN bits: WMMA propagates NaN thru x0)
            int r0 = rbase + k0; if (r0 > DH - 1) r0 = DH - 1;
            int r1 = rbase + k1; if (r1 > DH - 1) r1 = DH - 1;
            v2f bv;
            bv.x = sH[r0 * HP + colw];
            bv.y = sH[r1 * HP + colw];
            acc = __builtin_amdgcn_wmma_f32_16x16x4_f32(
                      /*neg_a=*/false, a, /*neg_b=*/false, bv,
                      /*c_mod=*/(short)0, acc,
                      /*reuse_a=*/false, /*reuse_b=*/false);
        }

        // D layout: VGPR i -> M = i + half*8, N = n
        #pragma unroll
        for (int i = 0; i < 8; ++i) {
            const int m  = i + half * 8;
            const int gy = R0 + sy * 16 + m;
            const int gx = C0 + sx * 16 + n;
            out[(size_t)b * plane + (size_t)gy * IMG_W + gx] = acc[i] * inv225;
        }
    }
}

extern "C" void kernel_launch(void* const* d_in, const int* in_sizes, int n_in,
                              void* d_out, int out_size, void* d_ws, size_t ws_size,
                              hipStream_t stream) {
    const float* img = (const float*)d_in[0];
    // d_in[1] is the all-ones 15x15 weight; reference scales it by 1/225,
    // which is folded into the kernel analytically.
    float* out = (float*)d_out;

    dim3 grid(IMG_W / TX, IMG_H / TY, 16);   // 8 x 16 x 16 blocks
    dim3 block(256);                          // 8 waves (wave32)
    hipLaunchKernelGGL(dark_channel_box15, grid, block, 0, stream, img, out);

    (void)in_sizes; (void)n_in; (void)out_size; (void)d_ws; (void)ws_size;
}